// Restormer_74586402062592
// MI455X (gfx1250) — compile-verified
//
#include <hip/hip_runtime.h>
#include <math.h>

typedef float v2f __attribute__((ext_vector_type(2)));
typedef float v8f __attribute__((ext_vector_type(8)));

#define NCH   512
#define CT    768
#define NT    32
#define NPIX  (512*512)          // 262144
#define W1N   288                // 3 proj * 3 dx * 32 o
#define HEADN 256                // per-head spatial (16*16)

// ---------------------------------------------------------------------------
// K0: shifted/padded concatenation of q/k/v w1 -> W1cat[512][288]
// column n = p*96 + dx*32 + o ; W1cat[w][n] = w1_p[w-(dx-1)][o] (zero padded)
// ---------------------------------------------------------------------------
__global__ void build_w1cat_kernel(const float* __restrict__ qw1,
                                   const float* __restrict__ kw1,
                                   const float* __restrict__ vw1,
                                   float* __restrict__ W1cat) {
    int idx = blockIdx.x * 256 + threadIdx.x;
    if (idx >= NCH * W1N) return;
    int w = idx / W1N, n = idx % W1N;
    int p = n / 96, q = n % 96, dx = q / 32, o = q & 31;
    int src = w - (dx - 1);
    const float* w1 = (p == 0) ? qw1 : (p == 1) ? kw1 : vw1;
    W1cat[idx] = (src >= 0 && src < NCH) ? w1[src * NT + o] : 0.f;
}

// sumw1[p*32+o] = sum_w w1_p[w][o]  (for the conv-bias contribution)
__global__ void build_sumw1_kernel(const float* __restrict__ qw1,
                                   const float* __restrict__ kw1,
                                   const float* __restrict__ vw1,
                                   float* __restrict__ sumw1) {
    int t = threadIdx.x;
    if (t >= 96) return;
    int p = t / 32, o = t & 31;
    const float* w1 = (p == 0) ? qw1 : (p == 1) ? kw1 : vw1;
    float s = 0.f;
    for (int w = 0; w < NCH; ++w) s += w1[w * NT + o];
    sumw1[t] = s;
}

// ---------------------------------------------------------------------------
// Generic fp32 WMMA GEMM:  C[M,N] = A[M,K] * B[K,N], batched over blockIdx.z.
// Block = 128 threads = 4 waves; each wave owns one 16x16 C tile of a 32x32
// macro-tile. Uses V_WMMA_F32_16X16X4_F32 (ISA 7.12 layouts).
// ---------------------------------------------------------------------------
__global__ __launch_bounds__(128)
void gemm_f32_wmma_kernel(const float* __restrict__ A, const float* __restrict__ B,
                          float* __restrict__ C, int K, int lda, int ldb, int ldc,
                          long sA, long sB, long sC) {
    long z = blockIdx.z;
    A += z * sA; B += z * sB; C += z * sC;
    int w = threadIdx.x >> 5;
    int l = threadIdx.x & 31;
    int half = l >> 4;             // 0: lanes 0-15, 1: lanes 16-31
    int r = l & 15;
    int m0 = blockIdx.x * 32 + 16 * (w >> 1);
    int n0 = blockIdx.y * 32 + 16 * (w & 1);

    v8f acc = {};
    for (int k0 = 0; k0 < K; k0 += 4) {
        int kk = k0 + 2 * half;
        // A 16x4 f32: lane (half,r): VGPR0=A[r][kk], VGPR1=A[r][kk+1]
        v2f a = *(const v2f*)(A + (size_t)(m0 + r) * lda + kk);
        // B 4x16 f32: lane (half,r): VGPR0=B[kk][n0+r], VGPR1=B[kk+1][n0+r]
        v2f b;
        b.x = B[(size_t)kk * ldb + n0 + r];
        b.y = B[(size_t)(kk + 1) * ldb + n0 + r];
        acc = __builtin_amdgcn_wmma_f32_16x16x4_f32(false, a, false, b,
                                                    (short)0, acc, false, false);
    }
    // C/D: VGPR i -> M = i + 8*half, N = r
    for (int i = 0; i < 8; ++i)
        C[(size_t)(m0 + i + 8 * half) * ldc + n0 + r] = acc[i];
}

// ---------------------------------------------------------------------------
// Fused: u[h,o] = sigmoid(conv(x) row contracted) from Z, then stage-2
// m = sigmoid(w2^T u + b2) via WMMA against LDS, stored head-split.
// Grid (768 channels, 3 projections), 128 threads, 64KB LDS.
// ---------------------------------------------------------------------------
__global__ __launch_bounds__(128)
void fused_u_stage2_kernel(const float* __restrict__ Z,
                           const float* __restrict__ c_w, const float* __restrict__ c_b,
                           const float* __restrict__ qb1, const float* __restrict__ kb1,
                           const float* __restrict__ vb1, const float* __restrict__ sumw1,
                           const float* __restrict__ qw2, const float* __restrict__ kw2,
                           const float* __restrict__ vw2,
                           const float* __restrict__ qb2, const float* __restrict__ kb2,
                           const float* __restrict__ vb2,
                           float* __restrict__ Qh, float* __restrict__ KhT,
                           float* __restrict__ Vh) {
    __shared__ float u_lds[NCH * NT];   // 512*32*4 = 64 KB

    const int chn = blockIdx.x;          // 0..767
    const int p   = blockIdx.y;          // 0=q 1=k 2=v
    const int tid = threadIdx.x;

    const float* b1 = (p == 0) ? qb1 : (p == 1) ? kb1 : vb1;
    const float* w2 = (p == 0) ? qw2 : (p == 1) ? kw2 : vw2;
    const float* b2 = (p == 0) ? qb2 : (p == 1) ? kb2 : vb2;

    float cw27[27];
#pragma unroll
    for (int i = 0; i < 27; ++i) cw27[i] = c_w[chn * 27 + i];   // uniform -> SGPR
    const float cbias = c_b[chn];

    // ---- stage 1: u[h][o] (each thread: one h and an octet of 8 o's) ----
    for (int it = tid; it < 2048; it += 128) {
        int h  = it >> 2;
        int o0 = (it & 3) * 8;
        float acc[8] = {0, 0, 0, 0, 0, 0, 0, 0};
#pragma unroll
        for (int ci = 0; ci < 3; ++ci) {
#pragma unroll
            for (int dy = 0; dy < 3; ++dy) {
                int row = h + dy - 1;
                if (row < 0 || row >= NCH) continue;   // SAME padding in H
                const float* zr = Z + (size_t)(ci * NCH + row) * W1N + p * 96 + o0;
#pragma unroll
                for (int dx = 0; dx < 3; ++dx) {
                    float wgt = cw27[(ci * 3 + dy) * 3 + dx];
                    float4 z0 = *(const float4*)(zr + dx * 32);
                    float4 z1 = *(const float4*)(zr + dx * 32 + 4);
                    acc[0] += wgt * z0.x; acc[1] += wgt * z0.y;
                    acc[2] += wgt * z0.z; acc[3] += wgt * z0.w;
                    acc[4] += wgt * z1.x; acc[5] += wgt * z1.y;
                    acc[6] += wgt * z1.z; acc[7] += wgt * z1.w;
                }
            }
        }
#pragma unroll
        for (int j = 0; j < 8; ++j) {
            int o = o0 + j;
            float pre = acc[j] + b1[o] + cbias * sumw1[p * 32 + o];
            u_lds[h * NT + o] = 1.f / (1.f + expf(-pre));
        }
    }
    __syncthreads();

    // ---- stage 2: m[a][b] = sigmoid( sum_h w2[h][a] * u[h][b] + b2[a] ) ----
    // 32x32 output = 2x2 WMMA tiles, one per wave; K = 512 from LDS.
    int wv = tid >> 5, l = tid & 31, half = l >> 4, r = l & 15;
    int ta = wv >> 1, tb = wv & 1;
    v8f acc = {};
    for (int k0 = 0; k0 < NCH; k0 += 4) {
        int kk = k0 + 2 * half;
        v2f a;   // A[M=a][K=h] = w2[h*32 + a]
        a.x = w2[kk * NT + 16 * ta + r];
        a.y = w2[(kk + 1) * NT + 16 * ta + r];
        v2f b;   // B[K=h][N=b] = u_lds[h*32 + b]
        b.x = u_lds[kk * NT + 16 * tb + r];
        b.y = u_lds[(kk + 1) * NT + 16 * tb + r];
        acc = __builtin_amdgcn_wmma_f32_16x16x4_f32(false, a, false, b,
                                                    (short)0, acc, false, false);
    }
#pragma unroll
    for (int i = 0; i < 8; ++i) {
        int a_idx = 16 * ta + i + 8 * half;   // o2 (row of m)
        int b_idx = 16 * tb + r;              // n  (col of m)
        float mval = 1.f / (1.f + expf(-(acc[i] + b2[a_idx])));
        // head split: row = 2r+h1, col = 2c+h2 ; head = h1*2+h2 ; n = r*16+c
        int h1 = a_idx & 1, rr = a_idx >> 1;
        int h2 = b_idx & 1, cc = b_idx >> 1;
        int hd = h1 * 2 + h2;
        int n  = rr * 16 + cc;
        if (p == 0)      Qh [((size_t)hd * CT + chn) * HEADN + n]      = mval;
        else if (p == 1) KhT[((size_t)hd * HEADN + n) * CT + chn]      = mval;  // transposed
        else             Vh [((size_t)hd * CT + chn) * HEADN + n]      = mval;
    }
}

// ---------------------------------------------------------------------------
// Row softmax over S (4*768 rows of 768), with per-head temperature.
// One wave per row (24 elems/lane), wave32 shuffle reductions.
// ---------------------------------------------------------------------------
__global__ void softmax_rows_kernel(float* __restrict__ S, const float* __restrict__ temp) {
    int row  = blockIdx.x * 8 + (threadIdx.x >> 5);
    int lane = threadIdx.x & 31;
    float t = temp[row / CT];
    float* Rp = S + (size_t)row * CT;
    float vals[24];
    float mx = -3.4e38f;
#pragma unroll
    for (int i = 0; i < 24; ++i) {
        float v = Rp[lane + i * 32] * t;
        vals[i] = v;
        mx = fmaxf(mx, v);
    }
    for (int m = 16; m >= 1; m >>= 1) mx = fmaxf(mx, __shfl_xor(mx, m, 32));
    float s = 0.f;
#pragma unroll
    for (int i = 0; i < 24; ++i) { float e = expf(vals[i] - mx); vals[i] = e; s += e; }
    for (int m = 16; m >= 1; m >>= 1) s += __shfl_xor(s, m, 32);
    float inv = 1.f / s;
#pragma unroll
    for (int i = 0; i < 24; ++i) Rp[lane + i * 32] = vals[i] * inv;
}

// ---------------------------------------------------------------------------
// FFN: 1x1(3->12) -> dw3x3 -> exact-GELU gate -> 1x1(6->3), + residual.
// One thread per pixel.
// ---------------------------------------------------------------------------
__global__ void ffn_residual_kernel(const float* __restrict__ att, const float* __restrict__ x,
                                    const float* __restrict__ pi_w, const float* __restrict__ pi_b,
                                    const float* __restrict__ dw_w, const float* __restrict__ dw_b,
                                    const float* __restrict__ po_w, const float* __restrict__ po_b,
                                    float* __restrict__ out) {
    int pid = blockIdx.x * 256 + threadIdx.x;
    int H = pid >> 9, W = pid & 511;
    float z[12];
#pragma unroll
    for (int j = 0; j < 12; ++j) z[j] = dw_b[j];
    for (int dy = 0; dy < 3; ++dy) {
        int hh = H + dy - 1;
        if (hh < 0 || hh >= 512) continue;
        for (int dx = 0; dx < 3; ++dx) {
            int ww = W + dx - 1;
            if (ww < 0 || ww >= 512) continue;
            int np = hh * 512 + ww;
            float a0 = att[np], a1 = att[NPIX + np], a2 = att[2 * NPIX + np];
#pragma unroll
            for (int j = 0; j < 12; ++j) {
                float y = pi_b[j] + a0 * pi_w[3 * j] + a1 * pi_w[3 * j + 1] + a2 * pi_w[3 * j + 2];
                z[j] += dw_w[j * 9 + dy * 3 + dx] * y;
            }
        }
    }
    float g[6];
#pragma unroll
    for (int jj = 0; jj < 6; ++jj)
        g[jj] = 0.5f * z[jj] * (1.f + erff(z[jj] * 0.70710678f)) * z[jj + 6];
#pragma unroll
    for (int ch = 0; ch < 3; ++ch) {
        float acc = po_b[ch];
#pragma unroll
        for (int jj = 0; jj < 6; ++jj) acc += g[jj] * po_w[ch * 6 + jj];
        out[ch * NPIX + pid] = x[ch * NPIX + pid] + acc;
    }
}

// ---------------------------------------------------------------------------
extern "C" void kernel_launch(void* const* d_in, const int* in_sizes, int n_in,
                              void* d_out, int out_size, void* d_ws, size_t ws_size,
                              hipStream_t stream) {
    const float* x    = (const float*)d_in[0];
    const float* c_w  = (const float*)d_in[1];
    const float* c_b  = (const float*)d_in[2];
    const float* qw1  = (const float*)d_in[3];
    const float* qb1  = (const float*)d_in[4];
    const float* qw2  = (const float*)d_in[5];
    const float* qb2  = (const float*)d_in[6];
    const float* kw1  = (const float*)d_in[7];
    const float* kb1  = (const float*)d_in[8];
    const float* kw2  = (const float*)d_in[9];
    const float* kb2  = (const float*)d_in[10];
    const float* vw1  = (const float*)d_in[11];
    const float* vb1  = (const float*)d_in[12];
    const float* vw2  = (const float*)d_in[13];
    const float* vb2  = (const float*)d_in[14];
    const float* temp = (const float*)d_in[15];
    const float* pi_w = (const float*)d_in[16];
    const float* pi_b = (const float*)d_in[17];
    const float* dw_w = (const float*)d_in[18];
    const float* dw_b = (const float*)d_in[19];
    const float* po_w = (const float*)d_in[20];
    const float* po_b = (const float*)d_in[21];

    // workspace layout (floats)
    float* ws     = (float*)d_ws;
    float* W1cat  = ws;                       // 512*288        = 147456
    float* sumw1  = W1cat + 147456;           // 96
    float* Z      = sumw1 + 96;               // 1536*288       = 442368
    float* Qh     = Z + 442368;               // 4*768*256      = 786432
    float* KhT    = Qh + 786432;              // 4*256*768      = 786432
    float* Vh     = KhT + 786432;             // 4*768*256      = 786432
    float* S      = Vh + 786432;              // 4*768*768      = 2359296
    float* attimg = S + 2359296;              // 3*512*512      = 786432  (~24.4 MB total)

    // 0) shifted w1 concat + column sums
    build_w1cat_kernel<<<576, 256, 0, stream>>>(qw1, kw1, vw1, W1cat);
    build_sumw1_kernel<<<1, 96, 0, stream>>>(qw1, kw1, vw1, sumw1);

    // 1) Z[1536,288] = X[1536,512] * W1cat[512,288]   (WMMA)
    gemm_f32_wmma_kernel<<<dim3(48, 9, 1), 128, 0, stream>>>(
        x, W1cat, Z, 512, 512, W1N, W1N, 0, 0, 0);

    // 2) fused conv-contraction + sigmoid + stage-2 MLP (WMMA) + head split
    fused_u_stage2_kernel<<<dim3(CT, 3), 128, 0, stream>>>(
        Z, c_w, c_b, qb1, kb1, vb1, sumw1, qw2, kw2, vw2, qb2, kb2, vb2,
        Qh, KhT, Vh);

    // 3) S[hd] = Q[hd] * K[hd]^T  (M=768,N=768,K=256, batched over 4 heads)
    gemm_f32_wmma_kernel<<<dim3(24, 24, 4), 128, 0, stream>>>(
        Qh, KhT, S, 256, HEADN, CT, CT,
        (long)CT * HEADN, (long)HEADN * CT, (long)CT * CT);

    // 4) softmax (with temperature) over rows of S
    softmax_rows_kernel<<<384, 256, 0, stream>>>(S, temp);

    // 5) att[hd] = P[hd] * V[hd]  (M=768,N=256,K=768) -> directly (3,512,512)
    gemm_f32_wmma_kernel<<<dim3(24, 8, 4), 128, 0, stream>>>(
        S, Vh, attimg, CT, CT, HEADN, HEADN,
        (long)CT * CT, (long)CT * HEADN, (long)CT * HEADN);

    // 6) FFN + residual
    ffn_residual_kernel<<<1024, 256, 0, stream>>>(
        attimg, x, pi_w, pi_b, dw_w, dw_b, po_w, po_b, (float*)d_out);
}